// ChebNet_40063454937588
// MI455X (gfx1250) — compile-verified
//
#include <hip/hip_runtime.h>

typedef __attribute__((ext_vector_type(2))) float v2f;
typedef __attribute__((ext_vector_type(8))) float v8f;

// ---------------------------------------------------------------------------
// Edge-weight preparation
// ---------------------------------------------------------------------------
__global__ void cheb_degree_kernel(const int* __restrict__ row, float* __restrict__ deg, int E) {
    int e = blockIdx.x * blockDim.x + threadIdx.x;
    if (e < E) atomicAdd(&deg[row[e]], 1.0f);
}

__global__ void cheb_dis_kernel(float* __restrict__ deg, int n) {
    int i = blockIdx.x * blockDim.x + threadIdx.x;
    if (i < n) {
        float d = deg[i];
        deg[i] = (d > 0.0f) ? rsqrtf(fmaxf(d, 1.0f)) : 0.0f;
    }
}

__global__ void cheb_edgew_kernel(const int* __restrict__ row, const int* __restrict__ col,
                                  const float* __restrict__ dis, float* __restrict__ wE, int E) {
    int e = blockIdx.x * blockDim.x + threadIdx.x;
    if (e < E) wE[e] = -dis[row[e]] * dis[col[e]];
}

// T2 init: T2[i] = -T0[i]  (so T2 = 2*prop(T1) - T0 after the scatter)
__global__ void cheb_negcopy_kernel(const float* __restrict__ a, float* __restrict__ b, int n) {
    int i = blockIdx.x * blockDim.x + threadIdx.x;
    if (i < n) b[i] = -a[i];
}

// ---------------------------------------------------------------------------
// Sparse propagation: out[dst[e]] += scale * wE[e] * in[src[e]]
// 16 threads per edge, float4 (b128) gather + 4 fp32 atomics each.
// Edge scalars (src/dst/w) are identical across the 16 lanes -> one coalesced
// broadcast request. Gather and atomics each cover a contiguous 256B row.
// ---------------------------------------------------------------------------
__global__ void cheb_scatter_kernel(const float* __restrict__ in, float* __restrict__ out,
                                    const int* __restrict__ src, const int* __restrict__ dst,
                                    const float* __restrict__ wE, float scale, int E) {
    int t = blockIdx.x * blockDim.x + threadIdx.x;
    int e = t >> 4;
    if (e >= E) return;
    int c = t & 15;                       // 4-feature chunk within the 64-wide row
    int s = src[e];
    int d = dst[e];
    float w = wE[e] * scale;
    const float4 v = *(const float4*)(in + (size_t)s * 64 + (c << 2));
    float* op = out + (size_t)d * 64 + (c << 2);
    atomicAdd(op + 0, v.x * w);
    atomicAdd(op + 1, v.y * w);
    atomicAdd(op + 2, v.z * w);
    atomicAdd(op + 3, v.w * w);
}

// ---------------------------------------------------------------------------
// Fused Chebyshev GEMM:  out = [relu]( T0@W[0] + T1@W[1] + T2@W[2] + bias )
// One wave per (MT x 16)x16 output strip, fp32 WMMA 16x16x4. The B fragment
// (weight slice) is loaded once from LDS per (mat,kb) step and reused across
// MT M-tiles -> 5 WMMAs per 2 ds_load_b32.
//
// A frag (16x4 f32): lane L: row = L&15, VGPR0/1 = K = 4*kb + 2*(L>>4) + {0,1}
// B frag (4x16 f32): lane L: col = L&15, VGPR0/1 = K = 4*kb + 2*(L>>4) + {0,1}
// C/D (16x16 f32):   lane L: col = L&15, VGPR r = row (L>>4)*8 + r
// ---------------------------------------------------------------------------
template <int MT>
__global__ void __launch_bounds__(128)
cheb_gemm_wmma_kernel(const float* __restrict__ T0, const float* __restrict__ T1,
                      const float* __restrict__ T2, const float* __restrict__ W,
                      const float* __restrict__ bias, float* __restrict__ out,
                      int nTilesN, int F, int doRelu, int totalWaveTiles, int rowOffset) {
    __shared__ float sW[3 * 64 * 64 + 64];   // max layer: 3 x 64x64 weights + bias
    const int tid = threadIdx.y * 32 + threadIdx.x;
    const int nW = 3 * 64 * F;
    for (int i = tid; i < nW; i += 128) sW[i] = W[i];
    for (int i = tid; i < F; i += 128) sW[nW + i] = bias[i];
    __syncthreads();

    const int waveTile = blockIdx.x * 4 + threadIdx.y;   // wave-uniform
    if (waveTile >= totalWaveTiles) return;              // whole-wave exit: EXEC stays all-1s
    const int mGroup = waveTile / nTilesN;
    const int nTile = waveTile % nTilesN;

    const int lane = threadIdx.x;
    const int lg = lane >> 4;        // lane group (K half / row half)
    const int ln = lane & 15;
    const int rowBase = rowOffset + mGroup * (MT * 16);
    const int col = (nTile << 4) + ln;

    v8f acc[MT];
    const float bv = sW[nW + col];   // bias is per output column
#pragma unroll
    for (int i = 0; i < MT; ++i)
#pragma unroll
        for (int r = 0; r < 8; ++r) acc[i][r] = bv;

    const float* mats[3] = {T0, T1, T2};
#pragma unroll
    for (int m = 0; m < 3; ++m) {
        const float* __restrict__ T = mats[m];
        const float* Wm = &sW[m * 64 * F];
#pragma unroll
        for (int kb = 0; kb < 16; ++kb) {
            const int k0 = (kb << 2) + (lg << 1);
            v2f b;                                  // one B frag, reused MT times
            b.x = Wm[k0 * F + col];
            b.y = Wm[(k0 + 1) * F + col];
#pragma unroll
            for (int i = 0; i < MT; ++i) {
                v2f a = *(const v2f*)(T + (size_t)(rowBase + i * 16 + ln) * 64 + k0);
                acc[i] = __builtin_amdgcn_wmma_f32_16x16x4_f32(
                    false, a, false, b, (short)0, acc[i], false, false);
            }
        }
    }

#pragma unroll
    for (int i = 0; i < MT; ++i) {
        if (doRelu) {
#pragma unroll
            for (int r = 0; r < 8; ++r) acc[i][r] = fmaxf(acc[i][r], 0.0f);
        }
        float* orow = out + (size_t)(rowBase + i * 16 + (lg << 3)) * F + col;
#pragma unroll
        for (int r = 0; r < 8; ++r) orow[r * F] = acc[i][r];
    }
}

// ---------------------------------------------------------------------------
// Host-side layer driver
// ---------------------------------------------------------------------------
static void cheb_layer(const float* T0, const float* W, const float* b,
                       float* T1, float* T2, float* Hout,
                       const int* src, const int* dst, const float* wE,
                       int n, int E, int F, int relu, hipStream_t stream) {
    const int sthreads = E * 16;
    const int sblocks = (sthreads + 255) / 256;

    // T1 = prop(T0)
    hipMemsetAsync(T1, 0, (size_t)n * 64 * sizeof(float), stream);
    cheb_scatter_kernel<<<sblocks, 256, 0, stream>>>(T0, T1, src, dst, wE, 1.0f, E);

    // T2 = 2*prop(T1) - T0
    cheb_negcopy_kernel<<<(n * 64 + 255) / 256, 256, 0, stream>>>(T0, T2, n * 64);
    cheb_scatter_kernel<<<sblocks, 256, 0, stream>>>(T1, T2, src, dst, wE, 2.0f, E);

    // Hout = [relu](T0@W0 + T1@W1 + T2@W2 + b)
    const int nTilesN = F / 16;
    const int mTiles = n / 16;           // n divisible by 16 (50000 = 3125*16)
    const int mGroups = mTiles / 5;      // 5 M-tiles per wave
    const int wt5 = mGroups * nTilesN;
    if (wt5 > 0) {
        cheb_gemm_wmma_kernel<5><<<(wt5 + 3) / 4, dim3(32, 4), 0, stream>>>(
            T0, T1, T2, W, b, Hout, nTilesN, F, relu, wt5, 0);
    }
    const int remTiles = mTiles - mGroups * 5;   // 0 for N=50000; kept for shape safety
    if (remTiles > 0) {
        const int wt1 = remTiles * nTilesN;
        cheb_gemm_wmma_kernel<1><<<(wt1 + 3) / 4, dim3(32, 4), 0, stream>>>(
            T0, T1, T2, W, b, Hout, nTilesN, F, relu, wt1, mGroups * 80);
    }
}

extern "C" void kernel_launch(void* const* d_in, const int* in_sizes, int n_in,
                              void* d_out, int out_size, void* d_ws, size_t ws_size,
                              hipStream_t stream) {
    const float* x  = (const float*)d_in[0];
    const int*   ei = (const int*)d_in[1];
    const float* W1 = (const float*)d_in[2];
    const float* b1 = (const float*)d_in[3];
    const float* W2 = (const float*)d_in[4];
    const float* b2 = (const float*)d_in[5];
    const float* W3 = (const float*)d_in[6];
    const float* b3 = (const float*)d_in[7];

    const int n = in_sizes[0] / 64;   // 50000
    const int E = in_sizes[1] / 2;    // 800000
    const int* src = ei;              // edge_index[0]
    const int* dst = ei + E;          // edge_index[1]

    float* ws = (float*)d_ws;
    size_t o = 0;
    float* dis = ws + o; o += ((size_t)n + 63) & ~(size_t)63;
    float* wE  = ws + o; o += ((size_t)E + 63) & ~(size_t)63;
    float* T1  = ws + o; o += (size_t)n * 64;
    float* T2  = ws + o; o += (size_t)n * 64;
    float* HA  = ws + o; o += (size_t)n * 64;
    float* HB  = ws + o; o += (size_t)n * 64;

    // Edge weights: w = -rsqrt(deg[src]) * rsqrt(deg[dst])  (deg from src counts)
    hipMemsetAsync(dis, 0, (size_t)n * sizeof(float), stream);
    cheb_degree_kernel<<<(E + 255) / 256, 256, 0, stream>>>(src, dis, E);
    cheb_dis_kernel<<<(n + 255) / 256, 256, 0, stream>>>(dis, n);
    cheb_edgew_kernel<<<(E + 255) / 256, 256, 0, stream>>>(src, dst, dis, wE, E);

    // Three ChebConv layers
    cheb_layer(x,  W1, b1, T1, T2, HA, src, dst, wE, n, E, 64, 1, stream);
    cheb_layer(HA, W2, b2, T1, T2, HB, src, dst, wE, n, E, 64, 1, stream);
    cheb_layer(HB, W3, b3, T1, T2, (float*)d_out, src, dst, wE, n, E, 32, 0, stream);
}